// intra_att_LR_58102317580557
// MI455X (gfx1250) — compile-verified
//
#include <hip/hip_runtime.h>

#define N_NODES 50000
#define K_NEI   32
#define DIM     128

typedef float v2f __attribute__((ext_vector_type(2)));
typedef float v8f __attribute__((ext_vector_type(8)));

// out[n][i] = relu( sum_j X[n][j] * W[i][j] + b[i] )   (X:[n,128], W:[128,128] row-major)
// Per block: 16 output rows x 128 cols. X tile (16x128 = 8KB) is staged in LDS once via
// gfx1250 async-to-LDS loads; each of the 8 waves computes one 16x16 tile with
// V_WMMA_F32_16X16X4_F32, A from LDS, B from the L2-resident weight matrix.
__global__ __launch_bounds__(256) void gemm_bias_relu_wmma(
    const float* __restrict__ X, const float* __restrict__ W,
    const float* __restrict__ bias, float* __restrict__ out, int nrows)
{
    __shared__ __align__(16) float sX[16 * DIM];    // 8 KB X tile

    const int tid  = threadIdx.x;
    const int lane = tid & 31;
    const int wave = tid >> 5;
    const int row0 = blockIdx.x * 16;
    const int col0 = wave * 16;
    if (row0 >= nrows) return;

    // ---- async copy of the 16x128 X tile into LDS (512 x b128, 2 per thread) ----
    {
        const float*   Xblk = X + (size_t)row0 * DIM;
        const uint32_t lds0 = (uint32_t)(uintptr_t)(&sX[0]);   // low 32 bits = LDS offset
        #pragma unroll
        for (int i = tid; i < 512; i += 256) {
            uint32_t loff = lds0 + (uint32_t)i * 16u;
            uint64_t ga   = (uint64_t)(uintptr_t)(Xblk + i * 4);
            asm volatile("global_load_async_to_lds_b128 %0, %1, off"
                         :: "v"(loff), "v"(ga) : "memory");
        }
        asm volatile("s_wait_asynccnt 0" ::: "memory");
    }
    __syncthreads();

    const int m  = lane & 15;               // M (for A) / N (for B) index
    const int kh = (lane >> 4) << 1;        // K sub-offset: 0 (lanes 0-15) or 2 (lanes 16-31)

    // A lane layout (16x4 f32): v[r] = A[m][kk + kh + r]  -> contiguous float2 in LDS row
    // B lane layout (4x16 f32): v[r] = B[kh + r][m] = W[col0+m][kk + kh + r] -> contiguous float2
    const float* arow = &sX[m * DIM + kh];
    const float* brow = W + (size_t)(col0 + m) * DIM + kh;

    v8f acc = {0.f, 0.f, 0.f, 0.f, 0.f, 0.f, 0.f, 0.f};
    #pragma unroll 8
    for (int kk = 0; kk < DIM; kk += 4) {
        v2f a = *(const v2f*)(arow + kk);
        v2f b = *(const v2f*)(brow + kk);
        // (neg_a, A, neg_b, B, c_mod, C, reuse_a, reuse_b)
        acc = __builtin_amdgcn_wmma_f32_16x16x4_f32(false, a, false, b,
                                                    (short)0, acc, false, false);
    }

    // C/D layout: VGPR r -> row = r + 8*(lane/16), col = lane%16
    const int   colg  = col0 + m;
    const float bv    = bias[colg];
    const int   rbase = row0 + ((lane >> 4) << 3);
    #pragma unroll
    for (int r = 0; r < 8; ++r) {
        float v = acc[r] + bv;
        out[(size_t)(rbase + r) * DIM + colg] = v > 0.f ? v : 0.f;
    }
}

// One wave32 per node. lane k owns neighbor k (K==32==warpSize).
// Gathered hl rows are loaded ONCE (cooperative coalesced float4 loads) and cached in
// registers (32 x float4 = 128 VGPRs/lane), so the aggregation pass is pure VALU.
__global__ __launch_bounds__(256) void att_softmax_agg(
    const int* __restrict__ nei, const float* __restrict__ hl,
    const float* __restrict__ hr, const float* __restrict__ att,
    float* __restrict__ out)
{
    __shared__ float s_att[2 * DIM];        // [0,128)=a_ref  [128,256)=a_nei
    for (int i = threadIdx.x; i < 2 * DIM; i += blockDim.x) s_att[i] = att[i];
    __syncthreads();

    const int lane = threadIdx.x & 31;
    const int wave = threadIdx.x >> 5;
    const int n    = blockIdx.x * 8 + wave;
    if (n >= N_NODES) return;

    const float4 aref = *(const float4*)&s_att[4 * lane];
    const float4 anei = *(const float4*)&s_att[DIM + 4 * lane];
    const float4 hr4  = *(const float4*)&hr[(size_t)n * DIM + 4 * lane];

    // s_ref = dot(hr[n], a_ref)  (wave reduction)
    float sref = hr4.x * aref.x + hr4.y * aref.y + hr4.z * aref.z + hr4.w * aref.w;
    #pragma unroll
    for (int off = 16; off; off >>= 1) sref += __shfl_xor(sref, off, 32);

    const int myidx = nei[(size_t)n * K_NEI + lane];   // lane k's neighbor index

    // Pass 1: gather rows once, cache in registers, compute scores.
    float4 rows[K_NEI];
    float  sc = 0.f;
    #pragma unroll
    for (int k = 0; k < K_NEI; ++k) {
        const int idx = __shfl(myidx, k, 32);
        const float4 v = *(const float4*)&hl[(size_t)idx * DIM + 4 * lane];
        rows[k] = v;
        float p = v.x * anei.x + v.y * anei.y + v.z * anei.z + v.w * anei.w;
        #pragma unroll
        for (int off = 16; off; off >>= 1) p += __shfl_xor(p, off, 32);
        if (lane == k) sc = p;
    }

    // leaky_relu(sref + sc), then softmax across the 32 lanes
    float s = sref + sc;
    s = s > 0.f ? s : 0.01f * s;
    float mx = s;
    #pragma unroll
    for (int off = 16; off; off >>= 1) mx = fmaxf(mx, __shfl_xor(mx, off, 32));
    float e = __expf(s - mx);
    float sum = e;
    #pragma unroll
    for (int off = 16; off; off >>= 1) sum += __shfl_xor(sum, off, 32);
    const float w = e / sum;

    // Pass 2: agg from the register cache — zero memory traffic.
    float4 acc = make_float4(0.f, 0.f, 0.f, 0.f);
    #pragma unroll
    for (int k = 0; k < K_NEI; ++k) {
        const float wk = __shfl(w, k, 32);
        acc.x += wk * rows[k].x; acc.y += wk * rows[k].y;
        acc.z += wk * rows[k].z; acc.w += wk * rows[k].w;
    }

    float4 o;
    o.x = acc.x * hr4.x; o.y = acc.y * hr4.y; o.z = acc.z * hr4.z; o.w = acc.w * hr4.w;
    o.x = o.x > 0.f ? o.x : 0.f;
    o.y = o.y > 0.f ? o.y : 0.f;
    o.z = o.z > 0.f ? o.z : 0.f;
    o.w = o.w > 0.f ? o.w : 0.f;
    *(float4*)&out[(size_t)n * DIM + 4 * lane] = o;
}

extern "C" void kernel_launch(void* const* d_in, const int* in_sizes, int n_in,
                              void* d_out, int out_size, void* d_ws, size_t ws_size,
                              hipStream_t stream)
{
    const int*   nei     = (const int*)  d_in[0];
    const float* h       = (const float*)d_in[1];
    const float* h_refer = (const float*)d_in[2];
    const float* att     = (const float*)d_in[3];
    const float* Wl      = (const float*)d_in[4];
    const float* bl      = (const float*)d_in[5];
    const float* Wr      = (const float*)d_in[6];
    const float* br      = (const float*)d_in[7];
    float*       out     = (float*)d_out;

    float* hl = (float*)d_ws;                       // 50000*128 f32 = 25.6 MB
    float* hr = hl + (size_t)N_NODES * DIM;         // +25.6 MB

    const dim3 blk(256);
    const int gemm_blocks = (N_NODES + 15) / 16;    // 3125
    gemm_bias_relu_wmma<<<gemm_blocks, blk, 0, stream>>>(h,       Wl, bl, hl, N_NODES);
    gemm_bias_relu_wmma<<<gemm_blocks, blk, 0, stream>>>(h_refer, Wr, br, hr, N_NODES);

    const int att_blocks = (N_NODES + 7) / 8;       // 6250
    att_softmax_agg<<<att_blocks, blk, 0, stream>>>(nei, hl, hr, att, out);
}